// ExampleGNN_91319594647808
// MI455X (gfx1250) — compile-verified
//
#include <hip/hip_runtime.h>
#include <hip/hip_bf16.h>

typedef float v2f __attribute__((ext_vector_type(2)));
typedef float v8f __attribute__((ext_vector_type(8)));

#define DIM 128
#define OUTC 64
#define NGRAPHS 512

// ---------------------------------------------------------------------------
// BT[n*K + k] = B[k*N + n]  (one-time transpose of the small weight matrices
// so GEMM B-operands become contiguous 8-byte loads)
// ---------------------------------------------------------------------------
__global__ void transpose_kernel(const float* __restrict__ B,
                                 float* __restrict__ BT, int K, int N) {
    int idx = blockIdx.x * blockDim.x + threadIdx.x;
    if (idx >= K * N) return;
    int k = idx / N;
    int n = idx - k * N;
    BT[(size_t)n * K + k] = B[idx];
}

// ---------------------------------------------------------------------------
// C[M,N] = A[M,K] @ B[K,N] (+ bias[N]), B given TRANSPOSED as BT[N,K].
// One wave computes a 16x64 strip: 4 accumulators, A fragment reused 4x.
// Inner step: 5 contiguous b64 loads -> 4 V_WMMA_F32_16X16X4_F32.
// Requires M%16==0, N%64==0, K%4==0. blockDim.x = 256 (8 waves / block).
// ---------------------------------------------------------------------------
__global__ __launch_bounds__(256) void wmma_gemm_f32_bt(
    const float* __restrict__ A, const float* __restrict__ BT,
    const float* __restrict__ bias, float* __restrict__ C,
    int M, int N, int K)
{
    const int lane = threadIdx.x & 31;
    const int wave = blockIdx.x * (blockDim.x >> 5) + (threadIdx.x >> 5);
    const int colGroups = N >> 6;                 // 64-wide column groups
    const int tileM = wave / colGroups;
    const int cg    = wave - tileM * colGroups;
    if (tileM * 16 >= M) return;                  // wave-uniform: EXEC all-1s

    const int row0  = tileM << 4;
    const int col0  = cg << 6;
    const int m16   = lane & 15;                  // M offset (A) / N offset (B)
    const int khalf = (lane >> 4) << 1;           // lanes 0-15: K=0,1; 16-31: K=2,3

    const float* ap  = A  + (size_t)(row0 + m16) * K + khalf;
    const float* bp0 = BT + (size_t)(col0 + m16) * K + khalf;
    const float* bp1 = bp0 + (size_t)16 * K;
    const float* bp2 = bp0 + (size_t)32 * K;
    const float* bp3 = bp0 + (size_t)48 * K;

    v8f acc0 = {}, acc1 = {}, acc2 = {}, acc3 = {};
#pragma unroll 4
    for (int k0 = 0; k0 < K; k0 += 4) {
        v2f a  = *(const v2f*)(ap  + k0);
        v2f b0 = *(const v2f*)(bp0 + k0);
        v2f b1 = *(const v2f*)(bp1 + k0);
        v2f b2 = *(const v2f*)(bp2 + k0);
        v2f b3 = *(const v2f*)(bp3 + k0);
        acc0 = __builtin_amdgcn_wmma_f32_16x16x4_f32(false, a, false, b0, (short)0, acc0, false, false);
        acc1 = __builtin_amdgcn_wmma_f32_16x16x4_f32(false, a, false, b1, (short)0, acc1, false, false);
        acc2 = __builtin_amdgcn_wmma_f32_16x16x4_f32(false, a, false, b2, (short)0, acc2, false, false);
        acc3 = __builtin_amdgcn_wmma_f32_16x16x4_f32(false, a, false, b3, (short)0, acc3, false, false);
    }

    const int nOff  = lane & 15;
    const int mBase = row0 + ((lane >> 4) << 3);  // lanes 16-31 -> rows +8
    float* crow = C + (size_t)mBase * N + col0 + nOff;
#pragma unroll
    for (int r = 0; r < 8; ++r) {
        float b0v = bias ? bias[col0 + nOff]      : 0.0f;
        float b1v = bias ? bias[col0 + 16 + nOff] : 0.0f;
        float b2v = bias ? bias[col0 + 32 + nOff] : 0.0f;
        float b3v = bias ? bias[col0 + 48 + nOff] : 0.0f;
        crow[(size_t)r * N]      = acc0[r] + b0v;
        crow[(size_t)r * N + 16] = acc1[r] + b1v;
        crow[(size_t)r * N + 32] = acc2[r] + b2v;
        crow[(size_t)r * N + 48] = acc3[r] + b3v;
    }
}

// ---------------------------------------------------------------------------
// Degree / normalization helpers
// ---------------------------------------------------------------------------
__global__ void deg_init_kernel(float* __restrict__ deg, int n) {
    int i = blockIdx.x * blockDim.x + threadIdx.x;
    if (i < n) deg[i] = 1.0f;                     // self-loop
}

__global__ void deg_accum_kernel(const int* __restrict__ dst,
                                 float* __restrict__ deg, int e) {
    int i = blockIdx.x * blockDim.x + threadIdx.x;
    if (i < e)
        __hip_atomic_fetch_add(&deg[dst[i]], 1.0f,
                               __ATOMIC_RELAXED, __HIP_MEMORY_SCOPE_AGENT);
}

__global__ void dinv_kernel(const float* __restrict__ deg,
                            float* __restrict__ dinv, int n) {
    int i = blockIdx.x * blockDim.x + threadIdx.x;
    if (i < n) dinv[i] = rsqrtf(deg[i]);
}

// agg[i,c] = h[i,c]*dinv[i]^2 + bias[c]   (self-loop contribution + bias)
__global__ void self_bias_kernel(const float* __restrict__ h,
                                 const float* __restrict__ dinv,
                                 const float* __restrict__ bias,
                                 float* __restrict__ agg, int n) {
    int idx = blockIdx.x * blockDim.x + threadIdx.x;
    if (idx >= n * DIM) return;
    int i = idx >> 7;            // / 128
    int c = idx & (DIM - 1);
    float di = dinv[i];
    agg[idx] = h[idx] * di * di + bias[c];
}

// One wave per edge; each lane moves one float4 (4 channels) -> 512B coalesced
// gather of h[src], 4 hardware fp32 atomics into agg[dst].
__global__ __launch_bounds__(256) void edge_scatter_kernel(
    const float* __restrict__ h, const float* __restrict__ dinv,
    const int* __restrict__ src, const int* __restrict__ dst,
    float* __restrict__ agg, int n_edges)
{
    int wave = blockIdx.x * (blockDim.x >> 5) + (threadIdx.x >> 5);
    if (wave >= n_edges) return;                  // wave-uniform
    int lane = threadIdx.x & 31;
    int s = src[wave];
    int d = dst[wave];
    float coef = dinv[s] * dinv[d];
    const float4* hs = (const float4*)(h + (size_t)s * DIM);
    float4 v = hs[lane];
    float* out = agg + (size_t)d * DIM + lane * 4;
    __hip_atomic_fetch_add(&out[0], v.x * coef, __ATOMIC_RELAXED, __HIP_MEMORY_SCOPE_AGENT);
    __hip_atomic_fetch_add(&out[1], v.y * coef, __ATOMIC_RELAXED, __HIP_MEMORY_SCOPE_AGENT);
    __hip_atomic_fetch_add(&out[2], v.z * coef, __ATOMIC_RELAXED, __HIP_MEMORY_SCOPE_AGENT);
    __hip_atomic_fetch_add(&out[3], v.w * coef, __ATOMIC_RELAXED, __HIP_MEMORY_SCOPE_AGENT);
}

__global__ void relu_kernel(float* __restrict__ a, int total) {
    int i = blockIdx.x * blockDim.x + threadIdx.x;
    if (i < total) a[i] = fmaxf(a[i], 0.0f);
}

__global__ void zero_kernel(float* __restrict__ a, int total) {
    int i = blockIdx.x * blockDim.x + threadIdx.x;
    if (i < total) a[i] = 0.0f;
}

// pooled[batch[i], c] += h[i, c]
__global__ void pool_kernel(const float* __restrict__ h,
                            const int* __restrict__ batch,
                            float* __restrict__ pooled, int n) {
    int idx = blockIdx.x * blockDim.x + threadIdx.x;
    if (idx >= n * DIM) return;
    int i = idx >> 7;
    int c = idx & (DIM - 1);
    __hip_atomic_fetch_add(&pooled[(size_t)batch[i] * DIM + c], h[idx],
                           __ATOMIC_RELAXED, __HIP_MEMORY_SCOPE_AGENT);
}

// log_softmax over rows of 64; one wave32 per row, 2 columns per lane.
__global__ __launch_bounds__(256) void log_softmax_kernel(
    const float* __restrict__ logits, float* __restrict__ out, int rows)
{
    int wave = blockIdx.x * (blockDim.x >> 5) + (threadIdx.x >> 5);
    if (wave >= rows) return;
    int lane = threadIdx.x & 31;
    const float* r = logits + (size_t)wave * OUTC;
    float v0 = r[lane];
    float v1 = r[lane + 32];
    float m = fmaxf(v0, v1);
#pragma unroll
    for (int off = 16; off > 0; off >>= 1)
        m = fmaxf(m, __shfl_xor(m, off, 32));
    float s = __expf(v0 - m) + __expf(v1 - m);
#pragma unroll
    for (int off = 16; off > 0; off >>= 1)
        s += __shfl_xor(s, off, 32);
    float ls = __logf(s);
    out[(size_t)wave * OUTC + lane]      = v0 - m - ls;
    out[(size_t)wave * OUTC + lane + 32] = v1 - m - ls;
}

// ---------------------------------------------------------------------------
extern "C" void kernel_launch(void* const* d_in, const int* in_sizes, int n_in,
                              void* d_out, int out_size, void* d_ws, size_t ws_size,
                              hipStream_t stream) {
    const float* x       = (const float*)d_in[0];
    const int*   eidx    = (const int*)d_in[1];
    const int*   x_batch = (const int*)d_in[2];
    const float* W1      = (const float*)d_in[3];
    const float* b1      = (const float*)d_in[4];
    const float* W2      = (const float*)d_in[5];
    const float* b2      = (const float*)d_in[6];
    const float* Wh      = (const float*)d_in[7];
    const float* bh      = (const float*)d_in[8];
    float*       out     = (float*)d_out;

    const int n_nodes = in_sizes[0] / DIM;       // 50000
    const int n_edges = in_sizes[1] / 2;         // 640000
    const int* src = eidx;
    const int* dst = eidx + n_edges;

    // Workspace carve-up (fp32)
    float* deg    = (float*)d_ws;                          // n_nodes
    float* dinv   = deg    + n_nodes;                      // n_nodes
    float* h      = dinv   + n_nodes;                      // n_nodes*DIM
    float* agg    = h      + (size_t)n_nodes * DIM;        // n_nodes*DIM
    float* pooled = agg    + (size_t)n_nodes * DIM;        // NGRAPHS*DIM
    float* logits = pooled + (size_t)NGRAPHS * DIM;        // NGRAPHS*OUTC
    float* W1T    = logits + (size_t)NGRAPHS * OUTC;       // DIM*DIM
    float* W2T    = W1T    + DIM * DIM;                    // DIM*DIM
    float* WhT    = W2T    + DIM * DIM;                    // OUTC*DIM

    const int T = 256;
    const int nodeBlk = (n_nodes + T - 1) / T;
    const int edgeBlk = (n_edges + T - 1) / T;
    const int nfBlk   = (n_nodes * DIM + T - 1) / T;
    const int edgeWaveBlk = (n_edges + 7) / 8;   // 8 waves per block

    // One-time weight transposes (tiny)
    transpose_kernel<<<(DIM * DIM + T - 1) / T, T, 0, stream>>>(W1, W1T, DIM, DIM);
    transpose_kernel<<<(DIM * DIM + T - 1) / T, T, 0, stream>>>(W2, W2T, DIM, DIM);
    transpose_kernel<<<(DIM * OUTC + T - 1) / T, T, 0, stream>>>(Wh, WhT, DIM, OUTC);

    // Degree + symmetric normalization
    deg_init_kernel<<<nodeBlk, T, 0, stream>>>(deg, n_nodes);
    deg_accum_kernel<<<edgeBlk, T, 0, stream>>>(dst, deg, n_edges);
    dinv_kernel<<<nodeBlk, T, 0, stream>>>(deg, dinv, n_nodes);

    // ----- Layer 1: h = x @ W1 ; agg = scatter + self-loop + bias ; relu -----
    {
        int waves  = (n_nodes / 16) * (DIM / 64);
        int blocks = (waves + 7) / 8;
        wmma_gemm_f32_bt<<<blocks, T, 0, stream>>>(x, W1T, nullptr, h,
                                                   n_nodes, DIM, DIM);
    }
    self_bias_kernel<<<nfBlk, T, 0, stream>>>(h, dinv, b1, agg, n_nodes);
    edge_scatter_kernel<<<edgeWaveBlk, T, 0, stream>>>(h, dinv, src, dst, agg, n_edges);
    relu_kernel<<<nfBlk, T, 0, stream>>>(agg, n_nodes * DIM);

    // ----- Layer 2: h = relu(agg) @ W2 ; agg = scatter + self + bias ; relu --
    {
        int waves  = (n_nodes / 16) * (DIM / 64);
        int blocks = (waves + 7) / 8;
        wmma_gemm_f32_bt<<<blocks, T, 0, stream>>>(agg, W2T, nullptr, h,
                                                   n_nodes, DIM, DIM);
    }
    self_bias_kernel<<<nfBlk, T, 0, stream>>>(h, dinv, b2, agg, n_nodes);
    edge_scatter_kernel<<<edgeWaveBlk, T, 0, stream>>>(h, dinv, src, dst, agg, n_edges);
    relu_kernel<<<nfBlk, T, 0, stream>>>(agg, n_nodes * DIM);

    // ----- Pool -> dense -> log_softmax -----
    zero_kernel<<<(NGRAPHS * DIM + T - 1) / T, T, 0, stream>>>(pooled, NGRAPHS * DIM);
    pool_kernel<<<nfBlk, T, 0, stream>>>(agg, x_batch, pooled, n_nodes);
    {
        int waves  = (NGRAPHS / 16) * (OUTC / 64);   // 32 waves
        int blocks = (waves + 7) / 8;
        wmma_gemm_f32_bt<<<blocks, T, 0, stream>>>(pooled, WhT, bh, logits,
                                                   NGRAPHS, OUTC, DIM);
    }
    log_softmax_kernel<<<(NGRAPHS + 7) / 8, T, 0, stream>>>(logits, out, NGRAPHS);
}